// EdgeOrientationField_23759759081732
// MI455X (gfx1250) — compile-verified
//
#include <hip/hip_runtime.h>

typedef float v2f __attribute__((ext_vector_type(2)));
typedef float v8f __attribute__((ext_vector_type(8)));

#define B_ 4
#define N_ 8192
#define K_ 48
#define NF_ 16
#define DIST_EPS 0.1f
#define NORM_EPS 0.1f
#define QUAT_EPS 0.001f
#define FA_EPS   1e-5f

// ws layout (float offsets)
#define WS_SUMS  0        // [B][16][10]     = 640 floats (d=9 is the count)
#define WS_TABLE 1024     // [B][17][12]     = 816 floats (R row-major 9 + X 3)
#define WS_RNODE 2048     // [B*N][9]        = 294912 floats
#define WS_VNODE 296960   // [B*N][12] + 16  = 393232 floats (V rows, padded)

// ---------------- helpers ----------------

__device__ __forceinline__ float sgnf(float x) {
    return (x > 0.0f ? 1.0f : 0.0f) - (x < 0.0f ? 1.0f : 0.0f);
}

// ---------------- kernel 0: zero the field-sum accumulators ----------------

__global__ void eof_zero_sums(float* __restrict__ ws) {
    int t = blockIdx.x * blockDim.x + threadIdx.x;
    if (t < B_ * 16 * 10) ws[WS_SUMS + t] = 0.0f;
}

// ---------------- kernel 1: per-node frames + V rows ------------------------
// R[a*3+col] with cols (n1,n2,n3);  V = [normed(N-CA), normed(C-CA), CA, 1, 0, 0]

__global__ void eof_node_frames(const float* __restrict__ X, float* __restrict__ ws) {
    int t = blockIdx.x * blockDim.x + threadIdx.x;
    if (t >= B_ * N_) return;
    const float* xr = X + (size_t)t * 12;
    float Nx = xr[0], Ny = xr[1], Nz = xr[2];
    float Ax = xr[3], Ay = xr[4], Az = xr[5];
    float Cx = xr[6], Cy = xr[7], Cz = xr[8];
    float u1x = Nx - Ax, u1y = Ny - Ay, u1z = Nz - Az;
    float s1 = rsqrtf(u1x*u1x + u1y*u1y + u1z*u1z + NORM_EPS);
    float n1x = u1x*s1, n1y = u1y*s1, n1z = u1z*s1;          // n1 (no re-norm)
    float u2x = Cx - Ax, u2y = Cy - Ay, u2z = Cz - Az;
    float s2 = rsqrtf(u2x*u2x + u2y*u2y + u2z*u2z + NORM_EPS);
    float v2x = u2x*s2, v2y = u2y*s2, v2z = u2z*s2;
    float cx = n1y*v2z - n1z*v2y, cy = n1z*v2x - n1x*v2z, cz = n1x*v2y - n1y*v2x;
    float sc = rsqrtf(cx*cx + cy*cy + cz*cz + NORM_EPS);
    float n2x = cx*sc, n2y = cy*sc, n2z = cz*sc;
    float dx = n1y*n2z - n1z*n2y, dy = n1z*n2x - n1x*n2z, dz = n1x*n2y - n1y*n2x;
    float sd = rsqrtf(dx*dx + dy*dy + dz*dz + NORM_EPS);
    float n3x = dx*sd, n3y = dy*sd, n3z = dz*sd;

    float* R = ws + WS_RNODE + (size_t)t * 9;
    R[0] = n1x; R[1] = n2x; R[2] = n3x;
    R[3] = n1y; R[4] = n2y; R[5] = n3y;
    R[6] = n1z; R[7] = n2z; R[8] = n3z;

    float4* V = (float4*)(ws + WS_VNODE + (size_t)t * 12);   // 48B rows, 16B aligned
    V[0] = make_float4(n1x, n1y, n1z, v2x);
    V[1] = make_float4(v2y, v2z, Ax, Ay);
    V[2] = make_float4(Az, 1.0f, 0.0f, 0.0f);
}

// ---------------- kernel 2: field sums via V_WMMA_F32_16X16X4_F32 ----------
// sums[b][f][d] = sum_n onehot(C[b][n]==f+1) * V[b][n][d]
// Inner loop: b64 C load + 2 cndmask (A) + 2 lane-indexed V loads (B) + WMMA.
// Two independent accumulators hide WMMA latency (8 nodes / iteration).

__global__ void eof_field_sums_wmma(const int* __restrict__ C,
                                    float* __restrict__ ws) {
    const int lane = threadIdx.x & 31;
    const int half = lane >> 4;     // 0: lanes 0-15, 1: lanes 16-31
    const int li   = lane & 15;
    const int waveGlobal = (blockIdx.x * blockDim.x + threadIdx.x) >> 5;
    const int wavesPerBatch = 64;
    const int b = waveGlobal / wavesPerBatch;
    const int w = waveGlobal % wavesPerBatch;
    const int nodesPerWave = N_ / wavesPerBatch;   // 128
    const int nbase = w * nodesPerWave;

    const int*   Cb = C + b * N_;
    const float* Vb = ws + WS_VNODE + (size_t)b * N_ * 12;

    v8f acc0 = {};
    v8f acc1 = {};
    for (int s = 0; s < nodesPerWave; s += 8) {
        // A VGPR0 = K=0 (lanes 0-15) / K=2 (lanes 16-31); VGPR1 = K=1 / K=3 (B mirrors)
        const int p0 = nbase + s + 2 * half;        // group 0: nodes p0, p0+1
        const int p1 = p0 + 4;                      // group 1: nodes p1, p1+1
        v2f a0, b0, a1, b1;
        a0.x = (Cb[p0]     == li + 1) ? 1.0f : 0.0f;
        a0.y = (Cb[p0 + 1] == li + 1) ? 1.0f : 0.0f;
        a1.x = (Cb[p1]     == li + 1) ? 1.0f : 0.0f;
        a1.y = (Cb[p1 + 1] == li + 1) ? 1.0f : 0.0f;
        b0.x = Vb[(size_t)p0 * 12 + li];
        b0.y = Vb[(size_t)(p0 + 1) * 12 + li];
        b1.x = Vb[(size_t)p1 * 12 + li];
        b1.y = Vb[(size_t)(p1 + 1) * 12 + li];
        acc0 = __builtin_amdgcn_wmma_f32_16x16x4_f32(
            false, a0, false, b0, (short)0, acc0, false, false);
        acc1 = __builtin_amdgcn_wmma_f32_16x16x4_f32(
            false, a1, false, b1, (short)0, acc1, false, false);
    }

    // C/D layout: VGPR r, lanes 0-15: M=r, N=lane; lanes 16-31: M=r+8, N=lane-16
    if (li < 10) {
        float* sums = ws + WS_SUMS + (size_t)b * 16 * 10;
#pragma unroll
        for (int r = 0; r < 8; ++r) {
            int f = r + 8 * half;
            atomicAdd(&sums[f * 10 + li], acc0[r] + acc1[r]);
        }
    }
}

// ---------------- kernel 3: per-(batch,field) table ------------------------

__global__ void eof_build_table(float* __restrict__ ws) {
    int t = blockIdx.x * blockDim.x + threadIdx.x;
    if (t >= B_ * (NF_ + 1)) return;
    int b = t / (NF_ + 1);
    int c = t % (NF_ + 1);
    float* e = ws + WS_TABLE + (size_t)(b * (NF_ + 1) + c) * 12;
    if (c == 0) {        // C==0 -> onehot all zero -> averages are exactly 0
#pragma unroll
        for (int i = 0; i < 12; ++i) e[i] = 0.0f;
        return;
    }
    const float* s = ws + WS_SUMS + (size_t)(b * 16 + (c - 1)) * 10;
    float inv = 1.0f / (s[9] + FA_EPS);
    float a1x = s[0]*inv, a1y = s[1]*inv, a1z = s[2]*inv;
    float a2x = s[3]*inv, a2y = s[4]*inv, a2z = s[5]*inv;
    float xx  = s[6]*inv, xy  = s[7]*inv, xz  = s[8]*inv;
    float r1 = rsqrtf(a1x*a1x + a1y*a1y + a1z*a1z + NORM_EPS);
    float n1x = a1x*r1, n1y = a1y*r1, n1z = a1z*r1;          // field n1 IS re-normed
    float r2 = rsqrtf(a2x*a2x + a2y*a2y + a2z*a2z + NORM_EPS);
    float v2x = a2x*r2, v2y = a2y*r2, v2z = a2z*r2;
    float cx = n1y*v2z - n1z*v2y, cy = n1z*v2x - n1x*v2z, cz = n1x*v2y - n1y*v2x;
    float rc = rsqrtf(cx*cx + cy*cy + cz*cz + NORM_EPS);
    float n2x = cx*rc, n2y = cy*rc, n2z = cz*rc;
    float dx = n1y*n2z - n1z*n2y, dy = n1z*n2x - n1x*n2z, dz = n1x*n2y - n1y*n2x;
    float rd = rsqrtf(dx*dx + dy*dy + dz*dz + NORM_EPS);
    float n3x = dx*rd, n3y = dy*rd, n3z = dz*rd;
    e[0] = n1x; e[1] = n2x; e[2] = n3x;
    e[3] = n1y; e[4] = n2y; e[5] = n3y;
    e[6] = n1z; e[7] = n2z; e[8] = n3z;
    e[9] = xx;  e[10] = xy; e[11] = xz;
}

// ---------------- transformation features ----------------------------------

__device__ __forceinline__ void tfeat(float xix, float xiy, float xiz,
                                      float xjx, float xjy, float xjz,
                                      const float* __restrict__ Ri,
                                      const float* __restrict__ Rj,
                                      float* __restrict__ o) {
    float dx = xjx - xix, dy = xjy - xiy, dz = xjz - xiz;
    float L = sqrtf(dx*dx + dy*dy + dz*dz + DIST_EPS);
    float invL = 1.0f / L;
    float u0 = (Ri[0]*dx + Ri[3]*dy + Ri[6]*dz) * invL;
    float u1 = (Ri[1]*dx + Ri[4]*dy + Ri[7]*dz) * invL;
    float u2 = (Ri[2]*dx + Ri[5]*dy + Ri[8]*dz) * invL;
    float m[9];
#pragma unroll
    for (int r = 0; r < 3; ++r)
#pragma unroll
        for (int c = 0; c < 3; ++c)
            m[r*3 + c] = Ri[0*3 + r]*Rj[0*3 + c] + Ri[1*3 + r]*Rj[1*3 + c]
                       + Ri[2*3 + r]*Rj[2*3 + c];
    float t0 =  m[0] + m[4] + m[8];
    float t1 =  m[0] - m[4] - m[8];
    float t2 = -m[0] + m[4] - m[8];
    float t3 = -m[0] - m[4] + m[8];
    float g0 = 0.5f * sqrtf(fabsf(1.0f + t0) + QUAT_EPS);
    float g1 = 0.5f * sqrtf(fabsf(1.0f + t1) + QUAT_EPS);
    float g2 = 0.5f * sqrtf(fabsf(1.0f + t2) + QUAT_EPS);
    float g3 = 0.5f * sqrtf(fabsf(1.0f + t3) + QUAT_EPS);
    float q0 = g0;                       // sign(1) == 1
    float q1 = sgnf(m[7] - m[5]) * g1;   // R21 - R12
    float q2 = sgnf(m[2] - m[6]) * g2;   // R02 - R20
    float q3 = sgnf(m[3] - m[1]) * g3;   // R10 - R01
    float qn = rsqrtf(q0*q0 + q1*q1 + q2*q2 + q3*q3);
    o[0] = logf(L + DIST_EPS);
    o[1] = u0; o[2] = u1; o[3] = u2;
    o[4] = q0*qn; o[5] = q1*qn; o[6] = q2*qn; o[7] = q3*qn;
}

// ---------------- kernel 4: per-edge features (store-BW bound) --------------

__global__ void eof_edge_features(const float* __restrict__ X,
                                  const int* __restrict__ E,
                                  const int* __restrict__ C,
                                  const float* __restrict__ ws,
                                  float* __restrict__ out) {
    size_t gid = (size_t)blockIdx.x * blockDim.x + threadIdx.x;
    if (gid >= (size_t)B_ * N_ * K_) return;
    size_t bn = gid / K_;
    int b = (int)(bn / N_);
    int j = E[gid];
    size_t bj = (size_t)b * N_ + j;
    int ci = C[bn];
    int cj = C[bj];
    const float* Ri = ws + WS_RNODE + bn * 9;
    const float* Rj = ws + WS_RNODE + bj * 9;
    const float* Xi = X + bn * 12 + 3;    // CA atom
    const float* Xj = X + bj * 12 + 3;
    const float* Ti = ws + WS_TABLE + (size_t)(b * (NF_ + 1) + ci) * 12;
    const float* Tj = ws + WS_TABLE + (size_t)(b * (NF_ + 1) + cj) * 12;

    float o[24];
    tfeat(Ti[9], Ti[10], Ti[11], Tj[9], Tj[10], Tj[11], Ti, Tj, o);       // h_ff
    tfeat(Ti[9], Ti[10], Ti[11], Xj[0], Xj[1], Xj[2],   Ti, Rj, o + 8);   // h_fn
    tfeat(Xi[0], Xi[1], Xi[2],   Xj[0], Xj[1], Xj[2],   Ri, Rj, o + 16);  // h_nn
    float mask = (ci > 0 && cj > 0) ? 1.0f : 0.0f;

    float4* op = (float4*)(out + gid * 24);   // 96B/edge, 16B-aligned
#pragma unroll
    for (int q = 0; q < 6; ++q)
        op[q] = make_float4(o[4*q]*mask, o[4*q+1]*mask, o[4*q+2]*mask, o[4*q+3]*mask);
}

// ---------------- launch ----------------------------------------------------

extern "C" void kernel_launch(void* const* d_in, const int* in_sizes, int n_in,
                              void* d_out, int out_size, void* d_ws, size_t ws_size,
                              hipStream_t stream) {
    (void)in_sizes; (void)n_in; (void)out_size; (void)ws_size;
    const float* X = (const float*)d_in[0];
    const int*   E = (const int*)d_in[1];
    const int*   C = (const int*)d_in[2];
    float* ws  = (float*)d_ws;
    float* out = (float*)d_out;

    hipLaunchKernelGGL(eof_zero_sums,       dim3(3),   dim3(256), 0, stream, ws);
    hipLaunchKernelGGL(eof_node_frames,     dim3((B_*N_ + 255)/256), dim3(256), 0, stream, X, ws);
    hipLaunchKernelGGL(eof_field_sums_wmma, dim3(32),  dim3(256), 0, stream, C, ws);
    hipLaunchKernelGGL(eof_build_table,     dim3(1),   dim3(128), 0, stream, ws);
    size_t nedge = (size_t)B_ * N_ * K_;
    hipLaunchKernelGGL(eof_edge_features,   dim3((unsigned)((nedge + 255)/256)), dim3(256),
                       0, stream, X, E, C, ws, out);
}